// Model_22643067585068
// MI455X (gfx1250) — compile-verified
//
#include <hip/hip_runtime.h>
#include <hip/hip_bf16.h>
#include <math.h>

// ---------------------------------------------------------------------------
// E(3)-equivariant 3D CNN for gfx1250 (MI455X).
//
// EMB is nonzero ONLY at the 6 face-neighbor offsets (1.14136*exp(-0.45));
// the linear part only touches the center tap => every conv is a 7-point
// cross stencil. Layer5 + spatial sum collapses to
//   pred[co] = sum_t sum_ci K5[t][ci][co] * (T[ci] - faceSlice_t[ci]).
//
// Middle convs (56->64, K=7*56=392) are implicit GEMM on
// v_wmma_f32_16x16x4_f32 (wave32): block = one (d,h) row, 4 waves = 4 cout
// tiles, 3 M-tiles of 16 voxels. Input strips staged in LDS (channel stride
// padded to 58 dwords -> conflict-free ds_load_2addr_b64 A-fragments; staging
// uses GLOBAL_LOAD_ASYNC_TO_LDS_B64 when the toolchain exposes it). Weights
// pre-packed on device in WMMA B-fragment order -> one coalesced b64 per
// lane per k-step. Gate fused in epilogue through LDS, 2 threads/voxel,
// fast-exp transcendentals, b128-mergeable loads/stores.
//
// Per-batch ping-pong activations (24.8MB each) stay L2-resident (192MB L2);
// HBM sees ~x-in + tiny outputs -> the loop is XDL-fed, not BW-fed.
// Workspace: ~51.2 MB.
// ---------------------------------------------------------------------------

typedef float v2f __attribute__((ext_vector_type(2)));
typedef float v4f __attribute__((ext_vector_type(4)));
typedef float v8f __attribute__((ext_vector_type(8)));

#define NVOX (48*48*48)
#define ACTF ((size_t)NVOX*56)

#define EMB_F   0.7277633f     // 1.14136*exp(-0.45)
#define C0_F    0.28209479f    // 1/sqrt(4pi)   (== c1/sqrt(3))
#define C1_F    0.48860251f    // sqrt(3/(4pi))
#define C1S2_F  0.34549414f    // c1/sqrt(2)
#define A20_F   0.22360680f    // 1/sqrt(20)
#define A24_F   0.20412415f    // 1/sqrt(24)
// gate RMS-normalizers, E[phi(g)^2]^-1/2 over N(0,1) (analytic ~= ref's MC)
#define CEVEN_F 1.53347f
#define CODD_F  1.19137f
#define CGATE_F 1.84591f

#if defined(__has_builtin)
#  if __has_builtin(__builtin_amdgcn_global_load_async_to_lds_b64) && \
      __has_builtin(__builtin_amdgcn_s_wait_asynccnt)
#    define ASYNC_LDS 1
#  endif
#endif

#ifdef ASYNC_LDS
// Builtin signature (from hipcc diagnostic): src is a pointer to a 2x int
// vector in the global address space; dst mirrors it in LDS; then imm offset
// and imm cpol.
typedef int v2i_vs __attribute__((vector_size(8)));
typedef __attribute__((address_space(1))) v2i_vs* gptr_b64;
typedef __attribute__((address_space(3))) v2i_vs* lptr_b64;
#endif

// -------------------------- small helpers ----------------------------------

__device__ __forceinline__ float fast_sig(float x) { return 1.f / (1.f + __expf(-x)); }
__device__ __forceinline__ float gelu_t(float x) {   // tanh-approx gelu == x*sig(2z)
  return x * fast_sig(1.5957691f * (x + 0.044715f * x * x * x));
}
__device__ __forceinline__ float soft_odd_f(float x) { return (1.f - __expf(-x * x)) * x; }

__device__ __forceinline__ void tap_n(int t, float n[3]) {
  n[0] = n[1] = n[2] = 0.f;
  if (t > 0) n[(t - 1) >> 1] = (t & 1) ? -1.f : 1.f;
  // t: 1:-d 2:+d 3:-h 4:+h 5:-w 6:+w
}

__device__ __forceinline__ float epsn(int i, int m, const float n[3]) {
  // sum_j eps_{i,j,m} * n[j]
  if (i == 0 && m == 1) return -n[2];
  if (i == 0 && m == 2) return  n[1];
  if (i == 1 && m == 0) return  n[2];
  if (i == 1 && m == 2) return -n[0];
  if (i == 2 && m == 0) return -n[1];
  if (i == 2 && m == 1) return  n[0];
  return 0.f;
}

__device__ __forceinline__ void gate64(const float* y, float* o) {
#pragma unroll
  for (int c = 0; c < 16; c++) o[c] = gelu_t(y[c]) * CEVEN_F;
#pragma unroll
  for (int c = 0; c < 16; c++) o[16 + c] = soft_odd_f(y[16 + c]) * CODD_F;
  float g[8];
#pragma unroll
  for (int k = 0; k < 8; k++) g[k] = fast_sig(y[32 + k]) * CGATE_F;
#pragma unroll
  for (int k = 0; k < 8; k++)
#pragma unroll
    for (int m = 0; m < 3; m++)
      o[32 + k * 3 + m] = y[40 + k * 3 + m] * g[k];
}

// -------------------------- kernel builders --------------------------------

__global__ void zero_f32(float* p, int n) {
  int i = blockIdx.x * blockDim.x + threadIdx.x;
  if (i < n) p[i] = 0.f;
}

// Layer 0 kernel: K0[7][64]  (din = 1)
__global__ void build_k0(const float* __restrict__ tw0, const float* __restrict__ tw1,
                         const float* __restrict__ tw2, const float* __restrict__ lw0,
                         const float* __restrict__ lw1, float* __restrict__ K0) {
  int co = threadIdx.x;
  if (co >= 64) return;
  for (int t = 0; t < 7; t++) {
    float val = 0.f;
    if (t == 0) {
      if (co < 16) val = lw0[co];
      else if (co >= 32 && co < 40) val = lw1[co - 32];
    } else {
      float n[3]; tap_n(t, n);
      if (co < 16) val = EMB_F * C0_F * tw0[co];
      else if (co >= 32 && co < 40) val = EMB_F * C0_F * tw1[co - 32];
      else if (co >= 52) {
        int v = (co - 52) / 3, m = (co - 52) % 3;
        val = EMB_F * C1_F * tw2[v] * n[m];
      }
    }
    K0[t * 64 + co] = val;
  }
}

struct MidParams { const float* tw[12]; const float* lw[5]; };

// K(tap, ci, co) for middle layers (IR_GATED(56) -> IR_HID(64))
__device__ float kmid_value(int t, int ci, int co, const MidParams& P) {
  int iblk, u, ii = 0;
  if (ci < 16)      { iblk = 0; u = ci; }
  else if (ci < 32) { iblk = 1; u = ci - 16; }
  else if (ci < 44) { iblk = 2; u = (ci - 32) / 3; ii = (ci - 32) % 3; }
  else              { iblk = 3; u = (ci - 44) / 3; ii = (ci - 44) % 3; }
  int oblk, v, mm = 0;
  if (co < 16)      { oblk = 0; v = co; }
  else if (co < 32) { oblk = 1; v = co - 16; }
  else if (co < 40) { oblk = 2; v = co - 32; }
  else if (co < 52) { oblk = 3; v = (co - 40) / 3; mm = (co - 40) % 3; }
  else              { oblk = 4; v = (co - 52) / 3; mm = (co - 52) % 3; }

  if (t == 0) {  // linear self-interaction at center tap
    if (iblk == 0 && oblk == 0) return P.lw[0][u * 16 + v] * 0.25f;
    if (iblk == 0 && oblk == 2) return P.lw[1][u *  8 + v] * 0.25f;
    if (iblk == 1 && oblk == 1) return P.lw[2][u * 16 + v] * 0.25f;
    if (iblk == 2 && oblk == 3) return (ii == mm) ? P.lw[3][u * 4 + v] * 0.5f : 0.f;
    if (iblk == 3 && oblk == 4) return (ii == mm) ? P.lw[4][u * 4 + v] * 0.5f : 0.f;
    return 0.f;
  }
  float n[3]; tap_n(t, n);
  const float Ec0 = EMB_F * C0_F, Ec1 = EMB_F * C1_F, Ec1s2 = EMB_F * C1S2_F;
  if (iblk == 0 && oblk == 0) return Ec0 * A20_F * P.tw[0][u * 16 + v];
  if (iblk == 0 && oblk == 2) return Ec0 * A20_F * P.tw[1][u *  8 + v];
  if (iblk == 0 && oblk == 4) return Ec1 * A24_F * P.tw[2][u *  4 + v] * n[mm];
  if (iblk == 1 && oblk == 1) return Ec0 * A20_F * P.tw[3][u * 16 + v];
  if (iblk == 1 && oblk == 3) return Ec1 * A24_F * P.tw[4][u *  4 + v] * n[mm];
  if (iblk == 2 && oblk == 3) return (ii == mm) ? Ec0 * A24_F * P.tw[5][u * 4 + v] : 0.f;
  if (iblk == 2 && oblk == 1) return Ec0 * A20_F * P.tw[6][u * 16 + v] * n[ii];     // c1/sqrt3 == c0
  if (iblk == 2 && oblk == 4) return Ec1s2 * A24_F * P.tw[7][u * 4 + v] * epsn(ii, mm, n);
  if (iblk == 3 && oblk == 4) return (ii == mm) ? Ec0 * A24_F * P.tw[8][u * 4 + v] : 0.f;
  if (iblk == 3 && oblk == 0) return Ec0 * A20_F * P.tw[9][u * 16 + v] * n[ii];
  if (iblk == 3 && oblk == 2) return Ec0 * A20_F * P.tw[10][u * 8 + v] * n[ii];
  if (iblk == 3 && oblk == 3) return Ec1s2 * A24_F * P.tw[11][u * 4 + v] * epsn(ii, mm, n);
  return 0.f;
}

// Pack directly into WMMA B-fragment layout:
// KB[(t*14+kk)*4 + ct][lane][j] = K(t, kk*4 + (lane>>4)*2 + j, ct*16 + (lane&15))
__global__ void build_kmid(MidParams P, float* __restrict__ KB) {
  int idx = blockIdx.x * blockDim.x + threadIdx.x;
  if (idx >= 7 * 14 * 4 * 32 * 2) return;
  int j = idx & 1, lane = (idx >> 1) & 31, ct = (idx >> 6) & 3, q = idx >> 8;
  int t = q / 14, kk = q % 14;
  int ci = kk * 4 + ((lane >> 4) << 1) + j;
  int co = ct * 16 + (lane & 15);
  KB[idx] = kmid_value(t, ci, co, P);
}

// Layer 5 kernel: K5[7][56][8]
__global__ void build_k5(const float* __restrict__ tw0, const float* __restrict__ tw1,
                         const float* __restrict__ tw2, const float* __restrict__ tw3,
                         const float* __restrict__ lw0, const float* __restrict__ lw1,
                         float* __restrict__ K5) {
  int idx = blockIdx.x * blockDim.x + threadIdx.x;
  if (idx >= 7 * 56 * 8) return;
  int co = idx % 8, ci = (idx / 8) % 56, t = idx / 448;
  float val = 0.f;
  if (t == 0) {
    if (co == 0) { if (ci >= 16 && ci < 32) val = lw1[ci - 16] * 0.25f; }
    else if (ci < 16) val = lw0[ci * 7 + (co - 1)] * 0.25f;
  } else {
    float n[3]; tap_n(t, n);
    const float Ec0 = EMB_F * C0_F;
    if (co == 0) {
      if (ci >= 16 && ci < 32) val = Ec0 * A20_F * tw1[ci - 16];
      else if (ci >= 32 && ci < 44) {
        int u = (ci - 32) / 3, i = (ci - 32) % 3;
        val = Ec0 * A20_F * tw2[u] * n[i];
      }
    } else {
      if (ci < 16) val = Ec0 * A20_F * tw0[ci * 7 + (co - 1)];
      else if (ci >= 44) {
        int u = (ci - 44) / 3, i = (ci - 44) % 3;
        val = Ec0 * A20_F * tw3[u * 7 + (co - 1)] * n[i];
      }
    }
  }
  K5[idx] = val;
}

// -------------------------- compute kernels --------------------------------

// Layer 0: din=1 -> 64, fused gate -> 56.  One thread per voxel.
__global__ __launch_bounds__(256) void conv0_gate(const float* __restrict__ x,
                                                  const float* __restrict__ K0,
                                                  float* __restrict__ out) {
  int v = blockIdx.x * blockDim.x + threadIdx.x;
  if (v >= NVOX) return;
  int w = v % 48, h = (v / 48) % 48, d = v / 2304;
  float tv[7];
  tv[0] = x[v];
  tv[1] = (d > 0)  ? x[v - 2304] : 0.f;
  tv[2] = (d < 47) ? x[v + 2304] : 0.f;
  tv[3] = (h > 0)  ? x[v - 48]   : 0.f;
  tv[4] = (h < 47) ? x[v + 48]   : 0.f;
  tv[5] = (w > 0)  ? x[v - 1]    : 0.f;
  tv[6] = (w < 47) ? x[v + 1]    : 0.f;
  float y[64];
#pragma unroll
  for (int co = 0; co < 64; co++) {
    float a = 0.f;
#pragma unroll
    for (int t = 0; t < 7; t++) a += tv[t] * K0[t * 64 + co];
    y[co] = a;
  }
  float o[56];
  gate64(y, o);
  float* dst = (float*)__builtin_assume_aligned(out + (size_t)v * 56, 16);
#pragma unroll
  for (int c = 0; c < 56; c++) dst[c] = o[c];
}

// Middle layers: 56 -> 64 (fused gate -> 56) via v_wmma_f32_16x16x4_f32.
// Block = one (d,h) row of 48 voxels; 4 waves = 4 cout tiles; 3 M-tiles/wave.
__global__ __launch_bounds__(128) void conv_mid_gate(const float* __restrict__ in,
                                                     const float* __restrict__ KB,
                                                     float* __restrict__ out) {
  __shared__ float smem[5 * 50 * 58];   // 58000 B; strips, ch-stride padded to 58
  const int d = blockIdx.x / 48, h = blockIdx.x % 48;

  // ---- stage 5 input strips (center w=-1..48, +/-d, +/-h) into LDS ----
  for (int idx = threadIdx.x; idx < 5 * 50 * 28; idx += 128) {
    int s = idx / 1400, r = idx % 1400, vx = r / 28, c2 = r % 28;
    int dd = d, hh = h;
    if (s == 1) dd = d - 1; else if (s == 2) dd = d + 1;
    else if (s == 3) hh = h - 1; else if (s == 4) hh = h + 1;
    int w = vx - 1;
    float* lp = smem + (s * 50 + vx) * 58 + c2 * 2;
    if ((unsigned)dd < 48u && (unsigned)hh < 48u && (unsigned)w < 48u) {
      const float* gp = in + (size_t)(((dd * 48 + hh) * 48 + w) * 56 + c2 * 2);
#ifdef ASYNC_LDS
      __builtin_amdgcn_global_load_async_to_lds_b64((gptr_b64)gp, (lptr_b64)lp, 0, 0);
#else
      *(v2f*)lp = *(const v2f*)gp;
#endif
    } else {
      *(v2f*)lp = (v2f){0.f, 0.f};
    }
  }
#ifdef ASYNC_LDS
  __builtin_amdgcn_s_wait_asynccnt(0);
#endif
  __syncthreads();

  const int lane = threadIdx.x & 31;
  const int wvid = threadIdx.x >> 5;       // cout tile 0..3
  const int cb   = wvid * 16;
  const int mrow = lane & 15;
  const int kp   = (lane >> 4) << 1;       // K-pair base per A/B fragment layout
  v8f acc0 = {}, acc1 = {}, acc2 = {};

  const int stripTab[7] = {0, 1, 2, 3, 4, 0, 0};
  const int shTab[7]    = {0, 0, 0, 0, 0, -1, 1};
#pragma unroll
  for (int t = 0; t < 7; t++) {
    const float* aBase = smem + (stripTab[t] * 50 + 1 + shTab[t]) * 58;
    const float* bBase = KB + (size_t)((t * 14) * 4 + wvid) * 64 + lane * 2;
#pragma unroll
    for (int kk = 0; kk < 14; kk++) {
      v2f b  = *(const v2f*)(bBase + kk * 256);          // coalesced b64
      int ci = kk * 4 + kp;
      v2f x0 = *(const v2f*)(aBase + (mrow     ) * 58 + ci);
      v2f x1 = *(const v2f*)(aBase + (mrow + 16) * 58 + ci);
      v2f x2 = *(const v2f*)(aBase + (mrow + 32) * 58 + ci);
      acc0 = __builtin_amdgcn_wmma_f32_16x16x4_f32(false, x0, false, b, (short)0, acc0, false, false);
      acc1 = __builtin_amdgcn_wmma_f32_16x16x4_f32(false, x1, false, b, (short)0, acc1, false, false);
      acc2 = __builtin_amdgcn_wmma_f32_16x16x4_f32(false, x2, false, b, (short)0, acc2, false, false);
    }
  }
  __syncthreads();

  // ---- spill D tiles to LDS (stride 68: conflict-free + 16B-aligned rows) ----
  {
    int vb = (lane >> 4) << 3;          // lanes 16-31 hold M=8..15
    int nn = cb + (lane & 15);
#pragma unroll
    for (int r = 0; r < 8; r++) {
      smem[( 0 + vb + r) * 68 + nn] = acc0[r];
      smem[(16 + vb + r) * 68 + nn] = acc1[r];
      smem[(32 + vb + r) * 68 + nn] = acc2[r];
    }
  }
  __syncthreads();

  // ---- fused gate: 2 threads per voxel (even half / odd half) ----
  if (threadIdx.x < 96) {
    int vx = threadIdx.x >> 1, half = threadIdx.x & 1;
    const float* y = smem + vx * 68;
    float* dst = (float*)__builtin_assume_aligned(
        out + ((size_t)blockIdx.x * 48 + vx) * 56, 16);
    if (half == 0) {
#pragma unroll
      for (int c = 0; c < 16; c++) dst[c] = gelu_t(y[c]) * CEVEN_F;
      float g[4];
#pragma unroll
      for (int k = 0; k < 4; k++) g[k] = fast_sig(y[32 + k]) * CGATE_F;
#pragma unroll
      for (int k = 0; k < 4; k++)
#pragma unroll
        for (int m = 0; m < 3; m++)
          dst[32 + k * 3 + m] = y[40 + k * 3 + m] * g[k];
    } else {
#pragma unroll
      for (int c = 0; c < 16; c++) dst[16 + c] = soft_odd_f(y[16 + c]) * CODD_F;
      float g[4];
#pragma unroll
      for (int k = 0; k < 4; k++) g[k] = fast_sig(y[36 + k]) * CGATE_F;
#pragma unroll
      for (int k = 0; k < 4; k++)
#pragma unroll
        for (int m = 0; m < 3; m++)
          dst[44 + k * 3 + m] = y[52 + k * 3 + m] * g[k];
    }
  }
}

// Per-batch face/total reduction: acc[7][56] = {T, d0, d47, h0, h47, w0, w47}
__global__ __launch_bounds__(64) void reduce_faces(const float* __restrict__ act,
                                                   float* __restrict__ acc) {
  int row = blockIdx.x, d = row / 48, h = row % 48;
  int ci = threadIdx.x;
  if (ci >= 56) return;
  const float* base = act + (size_t)row * 48 * 56;
  float s = 0.f;
  for (int w = 0; w < 48; w++) s += base[w * 56 + ci];
  float w0v  = base[ci];
  float w47v = base[47 * 56 + ci];
  atomicAdd(&acc[0 * 56 + ci], s);
  if (d == 0)  atomicAdd(&acc[1 * 56 + ci], s);
  if (d == 47) atomicAdd(&acc[2 * 56 + ci], s);
  if (h == 0)  atomicAdd(&acc[3 * 56 + ci], s);
  if (h == 47) atomicAdd(&acc[4 * 56 + ci], s);
  atomicAdd(&acc[5 * 56 + ci], w0v);
  atomicAdd(&acc[6 * 56 + ci], w47v);
}

// pred[b][co] = sum_t sum_ci K5[t][ci][co]*(T - faceSlice(t)); final combine.
__global__ void finalize_k(const float* __restrict__ acc, const float* __restrict__ K5,
                           float* __restrict__ out) {
  __shared__ float pred[4][8];
  int tid = threadIdx.x;
  const int fb[7] = {0, 2, 1, 4, 3, 6, 5};  // tap -> excluded face bucket
  if (tid < 32) {
    int b = tid >> 3, co = tid & 7;
    const float* A = acc + b * 392;
    float p = 0.f;
    for (int t = 0; t < 7; t++)
      for (int ci = 0; ci < 56; ci++) {
        float s = A[ci] - (t ? A[fb[t] * 56 + ci] : 0.f);
        p += K5[(t * 56 + ci) * 8 + co] * s;
      }
    pred[b][co] = p;
  }
  __syncthreads();
  if (tid < 32) {
    int b = tid >> 3, c = tid & 7;
    float o;
    if (c == 0)      o =  pred[b][0] * pred[b][1];
    else if (c == 1) o = -pred[b][0] * pred[b][1];
    else             o =  pred[b][c];
    out[b * 8 + c] = o;
  }
}

// -------------------------- host orchestration -----------------------------
// d_in layout (insertion order: x, then per layer tp[] then lin[]):
//   [0]=x ; layer0: [1..3]=tp,[4..5]=lin ; layers1-4: 17 arrays each from [6];
//   layer5: [74..77]=tp,[78..79]=lin.  Workspace: ~51.2 MB.

extern "C" void kernel_launch(void* const* d_in, const int* in_sizes, int n_in,
                              void* d_out, int out_size, void* d_ws, size_t ws_size,
                              hipStream_t stream) {
  (void)in_sizes; (void)n_in; (void)out_size; (void)ws_size;
  const float* x = (const float*)d_in[0];
  float* ws   = (float*)d_ws;
  float* actA = ws;
  float* actB = ws + ACTF;
  float* K0   = ws + 2 * ACTF;            // 448   (pad 512)
  float* KB   = K0 + 512;                 // 4 * 100352 (B-fragment packed)
  float* K5   = KB + 4 * 100352;          // 3136  (pad 3200)
  float* acc  = K5 + 3200;                // 4 * 392
  float* out  = (float*)d_out;

  build_k0<<<1, 64, 0, stream>>>((const float*)d_in[1], (const float*)d_in[2],
                                 (const float*)d_in[3], (const float*)d_in[4],
                                 (const float*)d_in[5], K0);
  for (int l = 0; l < 4; l++) {
    MidParams P;
    int base = 6 + l * 17;
    for (int i = 0; i < 12; i++) P.tw[i] = (const float*)d_in[base + i];
    for (int i = 0; i < 5;  i++) P.lw[i] = (const float*)d_in[base + 12 + i];
    build_kmid<<<98, 256, 0, stream>>>(P, KB + (size_t)l * 100352);
  }
  build_k5<<<13, 256, 0, stream>>>((const float*)d_in[74], (const float*)d_in[75],
                                   (const float*)d_in[76], (const float*)d_in[77],
                                   (const float*)d_in[78], (const float*)d_in[79], K5);
  zero_f32<<<7, 256, 0, stream>>>(acc, 4 * 7 * 56);

  for (int b = 0; b < 4; b++) {           // batch elements independent; L2-resident
    conv0_gate<<<(NVOX + 255) / 256, 256, 0, stream>>>(x + (size_t)b * NVOX, K0, actA);
    conv_mid_gate<<<2304, 128, 0, stream>>>(actA, KB + 0 * 100352, actB);
    conv_mid_gate<<<2304, 128, 0, stream>>>(actB, KB + 1 * 100352, actA);
    conv_mid_gate<<<2304, 128, 0, stream>>>(actA, KB + 2 * 100352, actB);
    conv_mid_gate<<<2304, 128, 0, stream>>>(actB, KB + 3 * 100352, actA);
    reduce_faces<<<2304, 64, 0, stream>>>(actA, acc + b * 392);
  }
  finalize_k<<<1, 64, 0, stream>>>(acc, K5, out);
}